// GeometricExtractor_81733227642906
// MI455X (gfx1250) — compile-verified
//
#include <hip/hip_runtime.h>
#include <hip/hip_bf16.h>

#define B_   8
#define N_   4096
#define KNB  9
#define TOPK 10
#define R_   (B_ * N_ * KNB)   // 294912 rows into the MLP

typedef __attribute__((ext_vector_type(2))) float v2f;
typedef __attribute__((ext_vector_type(8))) float v8f;

// ---------------------------------------------------------------------------
// Kernel 1: brute-force kNN via V_WMMA_F32_16X16X4_F32 Gram tiles.
// Block = 256 threads (8 waves), one batch, 128 queries per block.
// LDS packs the f32 operand halves so the lane's K-pair selection
// (x,y for lanes 0-15 / z,0 for lanes 16-31) is a pure address offset:
//   sA[0..N) = x   sA[N..2N) = z      (sB mirrors with y / 0)
// Hot loop per tile: 1 fused dual LDS load + 1 WMMA + 8 FMA + 7 MIN + 1 CMP;
// the sorted-insert chain runs only in a cold block when the tile's minimum
// distance beats the lane's current 10th-best.
// ---------------------------------------------------------------------------
__global__ __launch_bounds__(256) void knn_kernel(const float* __restrict__ X,
                                                  int* __restrict__ nbr)
{
    __shared__ float sA[2 * N_];
    __shared__ float sB[2 * N_];
    __shared__ float sX2[N_];          // 80 KB total of the 320 KB WGP LDS

    const int tid = threadIdx.x;
    const int b = blockIdx.x >> 5;                 // 32 blocks per batch
    const int qBase0 = (blockIdx.x & 31) * 128;

    for (int i = tid; i < N_; i += 256) {
        __builtin_prefetch(&X[(b * N_ + i + 512) * 3], 0, 0);   // global_prefetch_b8
        const float xx = X[(b * N_ + i) * 3 + 0];
        const float yy = X[(b * N_ + i) * 3 + 1];
        const float zz = X[(b * N_ + i) * 3 + 2];
        sA[i] = xx; sA[N_ + i] = zz;
        sB[i] = yy; sB[N_ + i] = 0.0f;
        sX2[i] = xx * xx + yy * yy + zz * zz;
    }
    __syncthreads();

    const int lane    = tid & 31;
    const int wave    = tid >> 5;
    const int halfSel = lane >> 4;                 // 0: K=0,1 lanes  1: K=2,3 lanes
    const int offH    = halfSel * N_;              // operand-half select by address
    const int q       = qBase0 + wave * 16 + (lane & 15);

    v2f bq;                                        // B operand (4x16 f32)
    bq[0] = sA[offH + q];
    bq[1] = sB[offH + q];
    const float x2q = sX2[q];

    float td[TOPK]; int ti[TOPK];
    #pragma unroll
    for (int s = 0; s < TOPK; ++s) { td[s] = 3.4e38f; ti[s] = -1; }

    // branchless sorted-insert; adjacent levels share their compares via CSE
    auto insert = [&](float d, int m) {
        #pragma unroll
        for (int s = TOPK - 1; s >= 1; --s) {
            const bool  lts  = d < td[s];
            const bool  ltsm = d < td[s - 1];
            const float nd   = ltsm ? td[s - 1] : d;
            const int   ni   = ltsm ? ti[s - 1] : m;
            td[s] = lts ? nd : td[s];
            ti[s] = lts ? ni : ti[s];
        }
        const bool lt0 = d < td[0];
        td[0] = lt0 ? d : td[0];
        ti[0] = lt0 ? m : ti[0];
    };

    auto doTile = [&](int t) {
        const int mA = t * 16 + (lane & 15);
        v2f a;                                     // A operand (16x4 f32)
        a[0] = sA[offH + mA];
        a[1] = sB[offH + mA];

        v8f c = {};
        c = __builtin_amdgcn_wmma_f32_16x16x4_f32(
                /*neg_a=*/false, a, /*neg_b=*/false, bq,
                /*c_mod=*/(short)0, c, /*reuse_a=*/false, /*reuse_b=*/false);

        const int mBase = t * 16 + halfSel * 8;    // uniform across the 16 lanes
        float xm[8];                               // contiguous -> ds_load_b128 x2
        #pragma unroll
        for (int j = 0; j < 8; ++j) xm[j] = sX2[mBase + j];

        float dv[8];
        #pragma unroll
        for (int j = 0; j < 8; ++j) dv[j] = fmaf(-2.0f, c[j], xm[j] + x2q);

        float mn01 = fminf(dv[0], dv[1]), mn23 = fminf(dv[2], dv[3]);
        float mn45 = fminf(dv[4], dv[5]), mn67 = fminf(dv[6], dv[7]);
        const float mn = fminf(fminf(mn01, mn23), fminf(mn45, mn67));

        if (mn < td[TOPK - 1]) {                   // single cold branch per tile
            #pragma unroll
            for (int j = 0; j < 8; ++j) insert(dv[j], mBase + j);
        }
    };

    for (int t = 0; t < N_ / 16; t += 2) {         // 2 WMMA issues per iteration
        doTile(t);
        doTile(t + 1);
    }

    // merge with half-wave partner (same query column, disjoint candidate rows)
    float od[TOPK]; int oi[TOPK];
    #pragma unroll
    for (int s = 0; s < TOPK; ++s) {
        od[s] = __shfl_xor(td[s], 16, 32);
        oi[s] = __shfl_xor(ti[s], 16, 32);
    }
    #pragma unroll
    for (int s = 0; s < TOPK; ++s) {
        if (od[s] < td[TOPK - 1]) insert(od[s], oi[s]);
    }

    if (lane < 16) {
        #pragma unroll
        for (int s = 0; s < KNB; ++s)              // drop slot 0 (self / min-dist)
            nbr[(b * N_ + q) * KNB + s] = ti[s + 1];
    }
}

// ---------------------------------------------------------------------------
// Kernel 2: geometric features. One thread per point: angular sort of the 9
// relative vectors (fully unrolled -> register arrays), then centroid /
// normal / position / angle / norms -> 10 features per (point, neighbor).
// ---------------------------------------------------------------------------
__global__ __launch_bounds__(256) void feat_kernel(const float* __restrict__ X,
                                                   const int* __restrict__ nbr,
                                                   float* __restrict__ feat)
{
    const int p = blockIdx.x * 256 + threadIdx.x;      // b*N + n
    if (p >= B_ * N_) return;

    const float cx = X[p * 3 + 0], cy = X[p * 3 + 1], cz = X[p * 3 + 2];
    const int batchBase = (p / N_) * N_;

    float rx[KNB], ry[KNB], rz[KNB], ph[KNB];
    #pragma unroll
    for (int j = 0; j < KNB; ++j) {
        const int nb = nbr[p * KNB + j];
        const int base = (batchBase + nb) * 3;
        rx[j] = X[base + 0] - cx;
        ry[j] = X[base + 1] - cy;
        rz[j] = X[base + 2] - cz;
        ph[j] = atan2f(ry[j], rx[j]);
    }

    #pragma unroll
    for (int a = 0; a < KNB - 1; ++a) {
        #pragma unroll
        for (int s = 0; s < KNB - 1 - a; ++s) {
            if (ph[s] > ph[s + 1]) {
                float t;
                t = ph[s]; ph[s] = ph[s + 1]; ph[s + 1] = t;
                t = rx[s]; rx[s] = rx[s + 1]; rx[s + 1] = t;
                t = ry[s]; ry[s] = ry[s + 1]; ry[s + 1] = t;
                t = rz[s]; rz[s] = rz[s + 1]; rz[s + 1] = t;
            }
        }
    }

    float mask;
    {
        const float nx0 = ry[0] * rz[1] - rz[0] * ry[1];
        mask = (nx0 > 0.0f) ? 1.0f : -1.0f;        // sign survives normalization
    }

    #pragma unroll
    for (int j = 0; j < KNB; ++j) {
        const int jn = (j + 1 == KNB) ? 0 : j + 1;
        const float ax = rx[j],  ay = ry[j],  az = rz[j];
        const float bx = rx[jn], by = ry[jn], bz = rz[jn];
        const float ccx = 0.5f * (ax + bx), ccy = 0.5f * (ay + by), ccz = 0.5f * (az + bz);
        float nx = ay * bz - az * by;
        float ny = az * bx - ax * bz;
        float nz = ax * by - ay * bx;
        const float nl  = sqrtf(nx * nx + ny * ny + nz * nz);
        const float inv = mask / (nl + 1e-6f);
        nx *= inv; ny *= inv; nz *= inv;
        const float pos  = (nx * ccx + ny * ccy + nz * ccz) * 0.57735026919f;
        const float dotv = ax * bx + ay * by + az * bz;
        const float nA   = sqrtf(ax * ax + ay * ay + az * az);
        const float nB   = sqrtf(bx * bx + by * by + bz * bz);
        float cosv = dotv / (nA * nB + 1e-8f);
        cosv = fminf(1.0f, fmaxf(-1.0f, cosv));
        const float ang = acosf(cosv);

        float* f = feat + (p * KNB + j) * 10;
        f[0] = ccx; f[1] = ccy; f[2] = ccz;
        f[3] = nx;  f[4] = ny;  f[5] = nz;
        f[6] = pos; f[7] = ang; f[8] = nA; f[9] = nB;
    }
}

// ---------------------------------------------------------------------------
// Kernel 3/5: (optional BN+ReLU of previous layer) + Linear(10->10) + partial
// BN statistics. Wave32 shfl_xor butterfly reduction, lane0 ds_add_f32 into
// LDS, then 20 global f32 atomics per block. Grids are exact (R_ % 256 == 0).
// In-place safe: each thread reads its full row before writing it.
// ---------------------------------------------------------------------------
__global__ __launch_bounds__(256) void lin_stats_kernel(
    const float* __restrict__ in, float* __restrict__ out,
    const float* __restrict__ W, const float* __restrict__ bias,
    const float* __restrict__ bnstats,   // mu[10], rsig[10] of previous layer
    const float* __restrict__ g, const float* __restrict__ be,
    float* __restrict__ gsum, float* __restrict__ gsq, int applyBN)
{
    __shared__ float sW[100], sb[10], sg[10], sbe[10], smu[10], srs[10];
    __shared__ float ssum[10], ssq[10];
    const int tid  = threadIdx.x;
    const int lane = tid & 31;
    if (tid < 100) sW[tid] = W[tid];
    if (tid < 10) {
        sb[tid] = bias[tid]; ssum[tid] = 0.0f; ssq[tid] = 0.0f;
        if (applyBN) {
            smu[tid] = bnstats[tid]; srs[tid] = bnstats[10 + tid];
            sg[tid] = g[tid]; sbe[tid] = be[tid];
        }
    }
    __syncthreads();

    const int r = blockIdx.x * 256 + tid;          // grid is exact, all lanes live
    float a[10], hv[10];
    #pragma unroll
    for (int i = 0; i < 10; ++i) {
        float v = in[r * 10 + i];
        if (applyBN) v = fmaxf((v - smu[i]) * srs[i] * sg[i] + sbe[i], 0.0f);
        a[i] = v;
    }
    #pragma unroll
    for (int o = 0; o < 10; ++o) {
        float h = sb[o];
        #pragma unroll
        for (int i = 0; i < 10; ++i) h = fmaf(a[i], sW[o * 10 + i], h);
        hv[o] = h;
        out[r * 10 + o] = h;
    }

    // wave32 butterfly reduction per channel; one LDS atomic per wave/channel
    #pragma unroll
    for (int o = 0; o < 10; ++o) {
        float v  = hv[o];
        float vq = hv[o] * hv[o];
        #pragma unroll
        for (int off = 16; off > 0; off >>= 1) {
            v  += __shfl_xor(v,  off, 32);
            vq += __shfl_xor(vq, off, 32);
        }
        if (lane == 0) {
            atomicAdd(&ssum[o], v);
            atomicAdd(&ssq[o],  vq);
        }
    }
    __syncthreads();
    if (tid < 10) {
        atomicAdd(&gsum[tid], ssum[tid]);
        atomicAdd(&gsq[tid],  ssq[tid]);
    }
}

__global__ void finalize_kernel(const float* __restrict__ gsum,
                                const float* __restrict__ gsq,
                                float* __restrict__ mu_rs)
{
    const int c = threadIdx.x;
    if (c < 10) {
        const float mu  = gsum[c] * (1.0f / (float)R_);
        const float var = gsq[c] * (1.0f / (float)R_) - mu * mu;
        mu_rs[c]      = mu;
        mu_rs[10 + c] = rsqrtf(var + 1e-5f);
    }
}

// ---------------------------------------------------------------------------
// Kernel 7: BN2 + ReLU + max over the 9 neighbors -> (B,N,10)
// ---------------------------------------------------------------------------
__global__ __launch_bounds__(256) void out_kernel(const float* __restrict__ h2,
                                                  const float* __restrict__ stats2,
                                                  const float* __restrict__ g2,
                                                  const float* __restrict__ be2,
                                                  float* __restrict__ out)
{
    const int p = blockIdx.x * 256 + threadIdx.x;
    if (p >= B_ * N_) return;

    float mu[10], rs[10], gg[10], bb[10], mx[10];
    #pragma unroll
    for (int c = 0; c < 10; ++c) {
        mu[c] = stats2[c]; rs[c] = stats2[10 + c];
        gg[c] = g2[c];     bb[c] = be2[c];
        mx[c] = -3.4e38f;
    }
    for (int j = 0; j < KNB; ++j) {
        #pragma unroll
        for (int c = 0; c < 10; ++c) {
            float v = h2[(p * KNB + j) * 10 + c];
            v = fmaxf((v - mu[c]) * rs[c] * gg[c] + bb[c], 0.0f);
            mx[c] = fmaxf(mx[c], v);
        }
    }
    #pragma unroll
    for (int c = 0; c < 10; ++c) out[p * 10 + c] = mx[c];
}

__global__ void zero_kernel(float* __restrict__ p, int n)
{
    const int i = blockIdx.x * 256 + threadIdx.x;
    if (i < n) p[i] = 0.0f;
}

// ---------------------------------------------------------------------------
extern "C" void kernel_launch(void* const* d_in, const int* in_sizes, int n_in,
                              void* d_out, int out_size, void* d_ws, size_t ws_size,
                              hipStream_t stream)
{
    (void)in_sizes; (void)n_in; (void)out_size; (void)ws_size;

    const float* X   = (const float*)d_in[0];
    const float* W1  = (const float*)d_in[1];
    const float* b1  = (const float*)d_in[2];
    const float* g1  = (const float*)d_in[3];
    const float* be1 = (const float*)d_in[4];
    const float* W2  = (const float*)d_in[5];
    const float* b2  = (const float*)d_in[6];
    const float* g2  = (const float*)d_in[7];
    const float* be2 = (const float*)d_in[8];
    float* out = (float*)d_out;

    float* ws   = (float*)d_ws;
    float* sum1 = ws + 0;
    float* sq1  = ws + 10;
    float* st1  = ws + 20;    // mu1[10], rsig1[10]
    float* sum2 = ws + 40;
    float* sq2  = ws + 50;
    float* st2  = ws + 60;    // mu2[10], rsig2[10]
    int*   nbr  = (int*)(ws + 80);           // R_ ints
    float* h    = ws + 80 + R_;              // R_*10 floats, reused feat -> h1 -> h2

    zero_kernel<<<1, 128, 0, stream>>>(ws, 80);
    knn_kernel<<<B_ * 32, 256, 0, stream>>>(X, nbr);
    feat_kernel<<<(B_ * N_) / 256, 256, 0, stream>>>(X, nbr, h);
    lin_stats_kernel<<<R_ / 256, 256, 0, stream>>>(h, h, W1, b1, nullptr, nullptr, nullptr,
                                                   sum1, sq1, 0);
    finalize_kernel<<<1, 32, 0, stream>>>(sum1, sq1, st1);
    lin_stats_kernel<<<R_ / 256, 256, 0, stream>>>(h, h, W2, b2, st1, g1, be1,
                                                   sum2, sq2, 1);
    finalize_kernel<<<1, 32, 0, stream>>>(sum2, sq2, st2);
    out_kernel<<<(B_ * N_) / 256, 256, 0, stream>>>(h, st2, g2, be2, out);
}